// Encoder_LaplaceGNN_PPISAGE_55714315764103
// MI455X (gfx1250) — compile-verified
//
#include <hip/hip_runtime.h>
#include <math.h>

typedef float v2f __attribute__((ext_vector_type(2)));
typedef float v8f __attribute__((ext_vector_type(8)));

#define DHID 128

// ---------------- degree / reciprocal ----------------
__global__ void degree_kernel(const int* __restrict__ dst, float* __restrict__ cnt, int E) {
    int e = blockIdx.x * blockDim.x + threadIdx.x;
    if (e < E) unsafeAtomicAdd(&cnt[dst[e]], 1.0f);
}

__global__ void rinv_kernel(float* __restrict__ cnt, int N) {
    int n = blockIdx.x * blockDim.x + threadIdx.x;
    if (n < N) cnt[n] = 1.0f / fmaxf(cnt[n], 1.0f);
}

// ---------------- edge scatter-add (mean numerator) ----------------
template <int C>
__global__ void scatter_kernel(const float* __restrict__ feat, const int* __restrict__ src,
                               const int* __restrict__ dst, float* __restrict__ agg,
                               long long total) {
    long long i = (long long)blockIdx.x * blockDim.x + threadIdx.x;
    if (i >= total) return;
    int e = (int)(i / C);
    int c = (int)(i % C);
    unsafeAtomicAdd(&agg[(long long)dst[e] * C + c], feat[(long long)src[e] * C + c]);
}

__global__ void scale_rows_kernel(float* __restrict__ agg, const float* __restrict__ rinv,
                                  int C, long long total) {
    long long i = (long long)blockIdx.x * blockDim.x + threadIdx.x;
    if (i >= total) return;
    agg[i] *= rinv[i / C];
}

// ---------------- WMMA GEMM: out = A1@W1 [+ A2@W2] [+ bias] [+ add1] [+ add2] ----------------
// A: [N,K] row-major, W: [K,128] row-major, out/add: [N,128].
// One wave per 16-row strip; 8 f32 16x16 accumulator tiles cover all 128 columns.
// K1/K2 compile-time; K2==0 -> no second matmul. Tail handled branchlessly via
// clamped addresses + select on the A fragment only (a==0 nullifies the product).
template <int K1, int K2>
__global__ void gemm_wmma_kernel(const float* __restrict__ A1, const float* __restrict__ W1,
                                 const float* __restrict__ A2, const float* __restrict__ W2,
                                 const float* __restrict__ bias,
                                 const float* __restrict__ add1, const float* __restrict__ add2,
                                 float* __restrict__ out, int N, int numRowTiles) {
    int lane = threadIdx.x & 31;
    int gw = blockIdx.x * (blockDim.x >> 5) + (threadIdx.x >> 5);
    if (gw >= numRowTiles) return;   // wave-uniform exit

    int m0   = gw * 16;
    int r15  = lane & 15;
    int half = lane >> 4;            // 0: lanes 0-15, 1: lanes 16-31
    int kh   = half * 2;             // A/B K sub-offset per ISA 16x4 / 4x16 f32 layout
    int row  = m0 + r15;
    int rowc = row < N ? row : N - 1;

    const v2f zero2 = {0.f, 0.f};
    v8f zero8 = {0.f, 0.f, 0.f, 0.f, 0.f, 0.f, 0.f, 0.f};
    v8f acc[8];
#pragma unroll
    for (int t = 0; t < 8; ++t) acc[t] = zero8;

#pragma unroll 2
    for (int k0 = 0; k0 < K1; k0 += 4) {
        int ka = k0 + kh;
        bool inr = (ka < K1);                // pair fully in/out (K even, ka even)
        int kc = inr ? ka : 0;               // clamped -> always a valid address
        v2f a = *(const v2f*)(A1 + (size_t)rowc * K1 + kc);
        a = inr ? a : zero2;                 // zero A kills the product for the pad
        const float* wp = W1 + kc * DHID + r15;
#pragma unroll
        for (int t = 0; t < 8; ++t) {
            v2f b;
            b.x = wp[t * 16];
            b.y = wp[DHID + t * 16];
            acc[t] = __builtin_amdgcn_wmma_f32_16x16x4_f32(
                false, a, false, b, (short)0, acc[t], false, false);
        }
    }
    if (K2 > 0) {
#pragma unroll 2
        for (int k0 = 0; k0 < K2; k0 += 4) {
            int ka = k0 + kh;
            bool inr = (ka < K2);
            int kc = inr ? ka : 0;
            v2f a = *(const v2f*)(A2 + (size_t)rowc * K2 + kc);
            a = inr ? a : zero2;
            const float* wp = W2 + kc * DHID + r15;
#pragma unroll
            for (int t = 0; t < 8; ++t) {
                v2f b;
                b.x = wp[t * 16];
                b.y = wp[DHID + t * 16];
                acc[t] = __builtin_amdgcn_wmma_f32_16x16x4_f32(
                    false, a, false, b, (short)0, acc[t], false, false);
            }
        }
    }

    // Epilogue. C/D layout: VGPR r -> M = r + 8*half, N = lane&15.
#pragma unroll
    for (int t = 0; t < 8; ++t) {
        int j = t * 16 + r15;
        float bv = bias ? bias[j] : 0.0f;
#pragma unroll
        for (int rr = 0; rr < 8; ++rr) {
            int n = m0 + half * 8 + rr;
            if (n < N) {
                long long idx = (long long)n * DHID + j;
                float v = acc[t][rr] + bv;
                if (add1) v += add1[idx];
                if (add2) v += add2[idx];
                out[idx] = v;
            }
        }
    }
}

// ---------------- graph-LayerNorm statistics (mean over all N*C elements) ----------------
__global__ void ln_stats_kernel(const float* __restrict__ h, long long count,
                                double* __restrict__ stats) {
    __shared__ double ssum[256];
    __shared__ double ssq[256];
    double s = 0.0, q = 0.0;
    long long stride = (long long)gridDim.x * blockDim.x;
    for (long long i = (long long)blockIdx.x * blockDim.x + threadIdx.x; i < count; i += stride) {
        double v = (double)h[i];
        s += v;
        q += v * v;
    }
    int tid = threadIdx.x;
    ssum[tid] = s;
    ssq[tid] = q;
    __syncthreads();
    for (int off = 128; off > 0; off >>= 1) {
        if (tid < off) {
            ssum[tid] += ssum[tid + off];
            ssq[tid]  += ssq[tid + off];
        }
        __syncthreads();
    }
    if (tid == 0) {
        unsafeAtomicAdd(&stats[0], ssum[0]);
        unsafeAtomicAdd(&stats[1], ssq[0]);
    }
}

__global__ void ln_final_kernel(const double* __restrict__ stats, float* __restrict__ msc,
                                double invCount) {
    double mean = stats[0] * invCount;
    double var  = stats[1] * invCount - mean * mean;
    double sd   = sqrt(var > 0.0 ? var : 0.0);
    msc[0] = (float)mean;
    msc[1] = (float)(1.0 / (sd + 1e-5));
}

// ---------------- fused normalize * affine + PReLU (in place) ----------------
__global__ void ln_prelu_kernel(float* __restrict__ h, const float* __restrict__ w,
                                const float* __restrict__ b, const float* __restrict__ a,
                                const float* __restrict__ msc, int C, long long total) {
    long long i = (long long)blockIdx.x * blockDim.x + threadIdx.x;
    if (i >= total) return;
    int c = (int)(i % C);
    float v = (h[i] - msc[0]) * msc[1];
    v = v * w[c] + b[c];
    float al = a[0];
    h[i] = v >= 0.0f ? v : al * v;
}

// ---------------- driver ----------------
extern "C" void kernel_launch(void* const* d_in, const int* in_sizes, int n_in,
                              void* d_out, int out_size, void* d_ws, size_t ws_size,
                              hipStream_t stream) {
    (void)n_in; (void)out_size; (void)ws_size;

    const float* x    = (const float*)d_in[0];
    const int*   esrc = (const int*)d_in[1];
    const int*   edst = (const int*)d_in[2];
    const float* Wl1 = (const float*)d_in[3];
    const float* Wr1 = (const float*)d_in[4];
    const float* b1  = (const float*)d_in[5];
    const float* Wl2 = (const float*)d_in[6];
    const float* Wr2 = (const float*)d_in[7];
    const float* b2  = (const float*)d_in[8];
    const float* Wl3 = (const float*)d_in[9];
    const float* Wr3 = (const float*)d_in[10];
    const float* b3  = (const float*)d_in[11];
    const float* Ws1 = (const float*)d_in[12];
    const float* Ws2 = (const float*)d_in[13];
    const float* lnw1 = (const float*)d_in[14];
    const float* lnb1 = (const float*)d_in[15];
    const float* lnw2 = (const float*)d_in[16];
    const float* lnb2 = (const float*)d_in[17];
    const float* lnw3 = (const float*)d_in[18];
    const float* lnb3 = (const float*)d_in[19];
    const float* a1 = (const float*)d_in[20];
    const float* a2 = (const float*)d_in[21];
    const float* a3 = (const float*)d_in[22];

    const int DIN = 50;
    const int N = in_sizes[0] / DIN;
    const int E = in_sizes[1];

    // Workspace layout (bytes): [0,16) double stats | [16,24) float mean/scale
    // | rinv N f32 | agg N*128 f32 | h1 N*128 f32 | inb N*128 f32
    char* ws = (char*)d_ws;
    double* stats = (double*)ws;
    float*  msc   = (float*)(ws + 16);
    float*  rinv  = (float*)(ws + 256);
    size_t off = 256 + (((size_t)N * 4 + 255) / 256) * 256;
    float* agg = (float*)(ws + off); off += (size_t)N * DHID * 4;
    float* h1  = (float*)(ws + off); off += (size_t)N * DHID * 4;
    float* inb = (float*)(ws + off); off += (size_t)N * DHID * 4;
    float* outp = (float*)d_out;   // also serves as the h2 buffer

    int numRowTiles = (N + 15) / 16;
    const int wavesPerBlock = 8;
    dim3 gemmBlock(32 * wavesPerBlock);
    int gemmGrid = (numRowTiles + wavesPerBlock - 1) / wavesPerBlock;

    long long NC = (long long)N * DHID;

#define LAUNCH_LN(buf, w_, b_, a_)                                                         \
    do {                                                                                   \
        hipMemsetAsync(stats, 0, 16, stream);                                              \
        ln_stats_kernel<<<1024, 256, 0, stream>>>(buf, NC, stats);                         \
        ln_final_kernel<<<1, 1, 0, stream>>>(stats, msc, 1.0 / (double)NC);                \
        ln_prelu_kernel<<<(int)((NC + 255) / 256), 256, 0, stream>>>(buf, w_, b_, a_, msc, \
                                                                     DHID, NC);            \
    } while (0)

    // Degree (shared across layers)
    hipMemsetAsync(rinv, 0, (size_t)N * 4, stream);
    degree_kernel<<<(E + 255) / 256, 256, 0, stream>>>(edst, rinv, E);
    rinv_kernel<<<(N + 255) / 256, 256, 0, stream>>>(rinv, N);

    // ---------- Layer 1: h1 = prelu(ln(mean(x)@Wl1 + x@Wr1 + b1)) ----------
    hipMemsetAsync(agg, 0, (size_t)N * DIN * 4, stream);
    {
        long long tot = (long long)E * DIN;
        scatter_kernel<50><<<(int)((tot + 255) / 256), 256, 0, stream>>>(x, esrc, edst, agg, tot);
        long long tn = (long long)N * DIN;
        scale_rows_kernel<<<(int)((tn + 255) / 256), 256, 0, stream>>>(agg, rinv, DIN, tn);
    }
    gemm_wmma_kernel<50, 50><<<gemmGrid, gemmBlock, 0, stream>>>(
        agg, Wl1, x, Wr1, b1, nullptr, nullptr, h1, N, numRowTiles);
    LAUNCH_LN(h1, lnw1, lnb1, a1);

    // ---------- Layer 2: in2 = h1 + x@Ws1 ; h2 = prelu(ln(sage(in2))) ----------
    gemm_wmma_kernel<50, 0><<<gemmGrid, gemmBlock, 0, stream>>>(
        x, Ws1, nullptr, nullptr, nullptr, h1, nullptr, inb, N, numRowTiles);
    hipMemsetAsync(agg, 0, (size_t)N * DHID * 4, stream);
    {
        long long tot = (long long)E * DHID;
        scatter_kernel<128><<<(int)((tot + 255) / 256), 256, 0, stream>>>(inb, esrc, edst, agg, tot);
        scale_rows_kernel<<<(int)((NC + 255) / 256), 256, 0, stream>>>(agg, rinv, DHID, NC);
    }
    gemm_wmma_kernel<128, 128><<<gemmGrid, gemmBlock, 0, stream>>>(
        agg, Wl2, inb, Wr2, b2, nullptr, nullptr, outp, N, numRowTiles);
    LAUNCH_LN(outp, lnw2, lnb2, a2);   // outp now holds h2

    // ---------- Layer 3: in3 = h1 + h2 + x@Ws2 ; ret = prelu(ln(sage(in3))) ----------
    gemm_wmma_kernel<50, 0><<<gemmGrid, gemmBlock, 0, stream>>>(
        x, Ws2, nullptr, nullptr, nullptr, h1, outp, inb, N, numRowTiles);
    hipMemsetAsync(agg, 0, (size_t)N * DHID * 4, stream);
    {
        long long tot = (long long)E * DHID;
        scatter_kernel<128><<<(int)((tot + 255) / 256), 256, 0, stream>>>(inb, esrc, edst, agg, tot);
        scale_rows_kernel<<<(int)((NC + 255) / 256), 256, 0, stream>>>(agg, rinv, DHID, NC);
    }
    gemm_wmma_kernel<128, 128><<<gemmGrid, gemmBlock, 0, stream>>>(
        agg, Wl3, inb, Wr3, b3, nullptr, nullptr, outp, N, numRowTiles);
    LAUNCH_LN(outp, lnw3, lnb3, a3);

#undef LAUNCH_LN
}